// GnnBody_70987219469117
// MI455X (gfx1250) — compile-verified
//
#include <hip/hip_runtime.h>

#define NH 50000
#define NV 100000
#define NE 150000
#define EDGES 300000
#define H 128
#define NLAYERS 3
#define NB 256
#define NSCAL 22
#define EMBED 128
#define POOLW (3 * H + NSCAL) /* 406 */

typedef float v2f __attribute__((ext_vector_type(2)));
typedef float v8f __attribute__((ext_vector_type(8)));

// ---------------------------------------------------------------- utilities
__global__ void fill_zero(float* __restrict__ p, int n) {
    int t = blockIdx.x * blockDim.x + threadIdx.x;
    if (t < n) p[t] = 0.0f;
}

// in-projection: out[node,col] = relu(sum_k x[node,k]*W[k,col] + b[col])
__global__ void proj_relu(const float* __restrict__ x, const float* __restrict__ W,
                          const float* __restrict__ b, float* __restrict__ out,
                          int n, int F) {
    int t = blockIdx.x * blockDim.x + threadIdx.x;
    if (t >= n * H) return;
    int node = t >> 7, col = t & (H - 1);
    const float* xr = x + node * F;
    float acc = b[col];
    for (int k = 0; k < F; ++k) acc = fmaf(xr[k], W[k * H + col], acc);
    out[t] = fmaxf(acc, 0.0f);
}

// per-destination in-degree (as float), then inverted
__global__ void edge_count(const int* __restrict__ dst, float* __restrict__ cnt, int e) {
    int t = blockIdx.x * blockDim.x + threadIdx.x;
    if (t < e) atomicAdd(&cnt[dst[t]], 1.0f);
}
__global__ void invert_cnt(float* __restrict__ c, int n) {
    int t = blockIdx.x * blockDim.x + threadIdx.x;
    if (t < n) c[t] = 1.0f / fmaxf(c[t], 1.0f);
}

// one wave(32) per edge: each lane moves 4 contiguous floats of the 128-dim row
__global__ void scatter_add(const float* __restrict__ xsrc, const int* __restrict__ src,
                            const int* __restrict__ dst, float* __restrict__ agg, int e) {
    int t = blockIdx.x * blockDim.x + threadIdx.x;
    int ed = t >> 5, lane = t & 31;
    if (ed >= e) return;
    int s = src[ed], d = dst[ed];
    const float4 v = *(const float4*)(xsrc + s * H + lane * 4);
    float* o = agg + d * H + lane * 4;
    atomicAdd(o + 0, v.x);
    atomicAdd(o + 1, v.y);
    atomicAdd(o + 2, v.z);
    atomicAdd(o + 3, v.w);
}

// ---------------------------------------------------------------- fused SAGE GEMM
// out[r,:] = f( (invc[r]*agg[r,:]) @ Wl + xdst[r,:] @ Wr + bias )
// mode 0: store raw        (first relation into a shared destination)
// mode 1: store relu(v)
// mode 2: store relu(0.5*(v + out_prev))   (HeteroConv mean of 2 relations)
// One wave computes a 16-row stripe across all 128 output cols using
// V_WMMA_F32_16X16X4_F32 (fp32 in/out, matches reference numerics).
__global__ void sage_gemm_wmma(const float* __restrict__ agg, const float* __restrict__ invc,
                               const float* __restrict__ xdst,
                               const float* __restrict__ Wl, const float* __restrict__ Wr,
                               const float* __restrict__ bias, float* __restrict__ out,
                               int n, int mode) {
    int gtid = blockIdx.x * blockDim.x + threadIdx.x;
    int wave = gtid >> 5;                 // uniform within the wave
    int lane = threadIdx.x & 31;
    int nRowTiles = (n + 15) >> 4;
    if (wave >= nRowTiles) return;        // wave-uniform exit: EXEC all-ones at WMMA
    int row0 = wave << 4;
    int m  = lane & 15;
    int kh = lane >> 4;                   // 0: K lanes {0,1}; 1: K lanes {2,3}
    int r = row0 + m;
    if (r >= n) r = n - 1;                // clamp loads; stores guarded below
    float sA = invc[r];

    v8f c[8];
#pragma unroll
    for (int j = 0; j < 8; ++j) c[j] = (v8f){0, 0, 0, 0, 0, 0, 0, 0};

    const float* arow = agg  + r * H;
    const float* xrow = xdst + r * H;

    // term 1: mean @ Wl  (inverse-degree scaling folded into the A fragment)
    for (int k = 0; k < H; k += 4) {
        int ka = k + (kh << 1);
        v2f a;
        a.x = arow[ka] * sA;
        a.y = arow[ka + 1] * sA;
#pragma unroll
        for (int j = 0; j < 8; ++j) {
            int col = (j << 4) + m;
            v2f bm;
            bm.x = Wl[ka * H + col];
            bm.y = Wl[(ka + 1) * H + col];
            c[j] = __builtin_amdgcn_wmma_f32_16x16x4_f32(false, a, false, bm,
                                                         (short)0, c[j], false, false);
        }
    }
    // term 2: x_dst @ Wr
    for (int k = 0; k < H; k += 4) {
        int ka = k + (kh << 1);
        v2f a;
        a.x = xrow[ka];
        a.y = xrow[ka + 1];
#pragma unroll
        for (int j = 0; j < 8; ++j) {
            int col = (j << 4) + m;
            v2f bm;
            bm.x = Wr[ka * H + col];
            bm.y = Wr[(ka + 1) * H + col];
            c[j] = __builtin_amdgcn_wmma_f32_16x16x4_f32(false, a, false, bm,
                                                         (short)0, c[j], false, false);
        }
    }

    // epilogue: bias + combine + relu; C/D layout: lanes 0-15 -> M=i, 16-31 -> M=i+8
#pragma unroll
    for (int j = 0; j < 8; ++j) {
        int col = (j << 4) + m;
        float bc = bias[col];
#pragma unroll
        for (int i = 0; i < 8; ++i) {
            int orow = row0 + i + (kh << 3);
            if (orow < n) {
                float v = c[j][i] + bc;
                float* op = out + orow * H + col;
                if (mode == 1)      v = fmaxf(v, 0.0f);
                else if (mode == 2) v = fmaxf(0.5f * (v + *op), 0.0f);
                *op = v;
            }
        }
    }
}

// ---------------------------------------------------------------- pooling + head
__global__ void pool_scatter(const float* __restrict__ x, const int* __restrict__ batch,
                             float* __restrict__ pooled, int n, int colbase) {
    int t = blockIdx.x * blockDim.x + threadIdx.x;
    int node = t >> 5, lane = t & 31;
    if (node >= n) return;
    int g = batch[node];
    const float4 v = *(const float4*)(x + node * H + lane * 4);
    float* p = pooled + g * POOLW + colbase + lane * 4;
    atomicAdd(p + 0, v.x);
    atomicAdd(p + 1, v.y);
    atomicAdd(p + 2, v.z);
    atomicAdd(p + 3, v.w);
}

__global__ void batch_count(const int* __restrict__ batch, float* __restrict__ cnt, int n) {
    int t = blockIdx.x * blockDim.x + threadIdx.x;
    if (t < n) atomicAdd(&cnt[batch[t]], 1.0f);
}

__global__ void pool_finalize(float* __restrict__ pooled, const float* __restrict__ cnt3,
                              const float* __restrict__ scalars) {
    int t = blockIdx.x * blockDim.x + threadIdx.x;
    if (t >= NB * POOLW) return;
    int g = t / POOLW, col = t % POOLW;
    if (col < 3 * H) {
        int type = col >> 7;
        pooled[t] = pooled[t] / fmaxf(cnt3[type * NB + g], 1.0f);
    } else {
        pooled[t] = scalars[g * NSCAL + (col - 3 * H)];
    }
}

__global__ void final_gemm(const float* __restrict__ pooled, const float* __restrict__ Wf,
                           const float* __restrict__ bf, float* __restrict__ out) {
    int t = blockIdx.x * blockDim.x + threadIdx.x;
    if (t >= NB * EMBED) return;
    int g = t >> 7, o = t & (EMBED - 1);
    float acc = bf[o];
    const float* pr = pooled + g * POOLW;
    for (int k = 0; k < POOLW; ++k) acc = fmaf(pr[k], Wf[k * EMBED + o], acc);
    out[t] = acc;
}

// ---------------------------------------------------------------- host driver
static inline unsigned cdiv(int n, int b) { return (unsigned)((n + b - 1) / b); }

extern "C" void kernel_launch(void* const* d_in, const int* in_sizes, int n_in,
                              void* d_out, int out_size, void* d_ws, size_t ws_size,
                              hipStream_t stream) {
    const float* x_hex   = (const float*)d_in[0];
    const float* x_vert  = (const float*)d_in[1];
    const float* x_edge  = (const float*)d_in[2];
    const float* scalars = (const float*)d_in[3];
    const int* src_hv = (const int*)d_in[4];
    const int* dst_hv = (const int*)d_in[5];
    const int* src_vh = (const int*)d_in[6];
    const int* dst_vh = (const int*)d_in[7];
    const int* src_ve = (const int*)d_in[8];
    const int* dst_ve = (const int*)d_in[9];
    const int* src_ev = (const int*)d_in[10];
    const int* dst_ev = (const int*)d_in[11];
    const int* batch_hex  = (const int*)d_in[12];
    const int* batch_vert = (const int*)d_in[13];
    const int* batch_edge = (const int*)d_in[14];
    const float* W_ph = (const float*)d_in[15];
    const float* b_ph = (const float*)d_in[16];
    const float* W_pv = (const float*)d_in[17];
    const float* b_pv = (const float*)d_in[18];
    const float* W_pe = (const float*)d_in[19];
    const float* b_pe = (const float*)d_in[20];
    const float* conv_Wl = (const float*)d_in[21];
    const float* conv_Wr = (const float*)d_in[22];
    const float* conv_b  = (const float*)d_in[23];
    const float* W_final = (const float*)d_in[24];
    const float* b_final = (const float*)d_in[25];
    float* out = (float*)d_out;
    (void)in_sizes; (void)n_in; (void)out_size; (void)ws_size;

    // workspace carve-up (floats)
    float* ws = (float*)d_ws;
    size_t off = 0;
    auto carve = [&](size_t n) { float* p = ws + off; off += n; return p; };
    float* hhA = carve((size_t)NH * H);
    float* hhB = carve((size_t)NH * H);
    float* hvA = carve((size_t)NV * H);
    float* hvB = carve((size_t)NV * H);
    float* heA = carve((size_t)NE * H);
    float* heB = carve((size_t)NE * H);
    float* agg = carve((size_t)NE * H);     // shared per-relation mean-sum buffer
    float* inv_hv = carve(NV);              // 1/deg for dst of hex->vert
    float* inv_vh = carve(NH);
    float* inv_ve = carve(NE);
    float* inv_ev = carve(NV);
    float* pooled = carve((size_t)NB * POOLW);
    float* bcnt   = carve(3 * NB);

    const int BS = 256;

    // --- input projections + relu
    proj_relu<<<cdiv(NH * H, BS), BS, 0, stream>>>(x_hex, W_ph, b_ph, hhA, NH, 16);
    proj_relu<<<cdiv(NV * H, BS), BS, 0, stream>>>(x_vert, W_pv, b_pv, hvA, NV, 10);
    proj_relu<<<cdiv(NE * H, BS), BS, 0, stream>>>(x_edge, W_pe, b_pe, heA, NE, 8);

    // --- layer-invariant inverse in-degrees, computed once
    fill_zero<<<cdiv(NV, BS), BS, 0, stream>>>(inv_hv, NV);
    fill_zero<<<cdiv(NH, BS), BS, 0, stream>>>(inv_vh, NH);
    fill_zero<<<cdiv(NE, BS), BS, 0, stream>>>(inv_ve, NE);
    fill_zero<<<cdiv(NV, BS), BS, 0, stream>>>(inv_ev, NV);
    edge_count<<<cdiv(EDGES, BS), BS, 0, stream>>>(dst_hv, inv_hv, EDGES);
    edge_count<<<cdiv(EDGES, BS), BS, 0, stream>>>(dst_vh, inv_vh, EDGES);
    edge_count<<<cdiv(EDGES, BS), BS, 0, stream>>>(dst_ve, inv_ve, EDGES);
    edge_count<<<cdiv(EDGES, BS), BS, 0, stream>>>(dst_ev, inv_ev, EDGES);
    invert_cnt<<<cdiv(NV, BS), BS, 0, stream>>>(inv_hv, NV);
    invert_cnt<<<cdiv(NH, BS), BS, 0, stream>>>(inv_vh, NH);
    invert_cnt<<<cdiv(NE, BS), BS, 0, stream>>>(inv_ve, NE);
    invert_cnt<<<cdiv(NV, BS), BS, 0, stream>>>(inv_ev, NV);

    float *hh = hhA, *hv = hvA, *he = heA;
    float *hhN = hhB, *hvN = hvB, *heN = heB;

    for (int l = 0; l < NLAYERS; ++l) {
        const float* Wl = conv_Wl + (size_t)l * 4 * H * H;
        const float* Wr = conv_Wr + (size_t)l * 4 * H * H;
        const float* bb = conv_b + (size_t)l * 4 * H;
        unsigned gV = cdiv(((NV + 15) / 16) * 32, BS);
        unsigned gH = cdiv(((NH + 15) / 16) * 32, BS);
        unsigned gE = cdiv(((NE + 15) / 16) * 32, BS);

        // rel 0: hex -> vert  (first writer of hvN, store raw)
        fill_zero<<<cdiv(NV * H, BS), BS, 0, stream>>>(agg, NV * H);
        scatter_add<<<cdiv(EDGES * 32, BS), BS, 0, stream>>>(hh, src_hv, dst_hv, agg, EDGES);
        sage_gemm_wmma<<<gV, BS, 0, stream>>>(agg, inv_hv, hv, Wl + 0 * H * H, Wr + 0 * H * H,
                                              bb + 0 * H, hvN, NV, 0);
        // rel 3: edge -> vert (second writer: relu(0.5*(prev+v)))
        fill_zero<<<cdiv(NV * H, BS), BS, 0, stream>>>(agg, NV * H);
        scatter_add<<<cdiv(EDGES * 32, BS), BS, 0, stream>>>(he, src_ev, dst_ev, agg, EDGES);
        sage_gemm_wmma<<<gV, BS, 0, stream>>>(agg, inv_ev, hv, Wl + 3 * H * H, Wr + 3 * H * H,
                                              bb + 3 * H, hvN, NV, 2);
        // rel 1: vert -> hex  (relu)
        fill_zero<<<cdiv(NH * H, BS), BS, 0, stream>>>(agg, NH * H);
        scatter_add<<<cdiv(EDGES * 32, BS), BS, 0, stream>>>(hv, src_vh, dst_vh, agg, EDGES);
        sage_gemm_wmma<<<gH, BS, 0, stream>>>(agg, inv_vh, hh, Wl + 1 * H * H, Wr + 1 * H * H,
                                              bb + 1 * H, hhN, NH, 1);
        // rel 2: vert -> edge (relu)
        fill_zero<<<cdiv(NE * H, BS), BS, 0, stream>>>(agg, NE * H);
        scatter_add<<<cdiv(EDGES * 32, BS), BS, 0, stream>>>(hv, src_ve, dst_ve, agg, EDGES);
        sage_gemm_wmma<<<gE, BS, 0, stream>>>(agg, inv_ve, he, Wl + 2 * H * H, Wr + 2 * H * H,
                                              bb + 2 * H, heN, NE, 1);

        float* t;
        t = hh; hh = hhN; hhN = t;
        t = hv; hv = hvN; hvN = t;
        t = he; he = heN; heN = t;
    }

    // --- mean pooling per graph + scalars concat + final linear head
    fill_zero<<<cdiv(NB * POOLW, BS), BS, 0, stream>>>(pooled, NB * POOLW);
    fill_zero<<<cdiv(3 * NB, BS), BS, 0, stream>>>(bcnt, 3 * NB);
    pool_scatter<<<cdiv(NH * 32, BS), BS, 0, stream>>>(hh, batch_hex, pooled, NH, 0);
    pool_scatter<<<cdiv(NV * 32, BS), BS, 0, stream>>>(hv, batch_vert, pooled, NV, H);
    pool_scatter<<<cdiv(NE * 32, BS), BS, 0, stream>>>(he, batch_edge, pooled, NE, 2 * H);
    batch_count<<<cdiv(NH, BS), BS, 0, stream>>>(batch_hex, bcnt + 0 * NB, NH);
    batch_count<<<cdiv(NV, BS), BS, 0, stream>>>(batch_vert, bcnt + 1 * NB, NV);
    batch_count<<<cdiv(NE, BS), BS, 0, stream>>>(batch_edge, bcnt + 2 * NB, NE);
    pool_finalize<<<cdiv(NB * POOLW, BS), BS, 0, stream>>>(pooled, bcnt, scalars);
    final_gemm<<<cdiv(NB * EMBED, BS), BS, 0, stream>>>(pooled, W_final, b_final, out);
}